// Mamba2Block_85925115724494
// MI455X (gfx1250) — compile-verified
//
#include <hip/hip_runtime.h>
#include <hip/hip_bf16.h>
#include <math.h>

typedef __attribute__((ext_vector_type(2))) float v2f;
typedef __attribute__((ext_vector_type(8))) float v8f;

#define BATCH   2
#define SEQLEN  1024
#define D_MODEL 128
#define D_INNER 256
#define D_STATE 256
#define D_CONV  4
#define NROWS   (BATCH * SEQLEN)          // 2048
#define XZ_N    (2 * D_INNER)             // 512
#define PROJ_N  (2 * D_STATE + D_INNER)   // 768

__device__ __forceinline__ float siluf(float x) {
    return x / (1.0f + __expf(-x));
}
__device__ __forceinline__ float softplusf(float x) {
    // numerically stable softplus: max(x,0) + log1p(exp(-|x|))
    return fmaxf(x, 0.0f) + log1pf(__expf(-fabsf(x)));
}

// ---------------------------------------------------------------------------
// fp32 GEMM using V_WMMA_F32_16X16X4_F32.
// One wave computes a 16 x (16*NT) strip: one A fragment per k-step feeds NT
// back-to-back WMMAs. Fragment loads are software-pipelined one k-step ahead
// so global loads overlap the WMMA group instead of stalling on loadcnt 0.
// A: MxK row-major, B: KxN row-major, D: MxN row-major.
// Requires M % (16*BM_WAVES) == 0, N % (16*NT) == 0, K % 4 == 0, K >= 8.
// ---------------------------------------------------------------------------
template <int BM_WAVES, int NT>
__global__ __launch_bounds__(32 * BM_WAVES) void gemm_f32_wmma(
    const float* __restrict__ A, const float* __restrict__ B,
    float* __restrict__ D, int M, int N, int K)
{
    const int lane = threadIdx.x & 31;
    const int wave = threadIdx.x >> 5;
    const int hi   = lane >> 4;   // 0: lanes 0-15, 1: lanes 16-31
    const int lo   = lane & 15;
    const int m0   = (blockIdx.y * BM_WAVES + wave) * 16;
    const int n0   = blockIdx.x * (16 * NT);
    if (m0 >= M) return;          // wave-uniform; EXEC stays all-ones for WMMA

    v8f acc[NT];
    #pragma unroll
    for (int i = 0; i < NT; ++i) acc[i] = (v8f){0.f,0.f,0.f,0.f,0.f,0.f,0.f,0.f};

    // A 16x4 f32 layout: lanes 0-15 hold K={k0,k0+1}, lanes 16-31 K={k0+2,k0+3}
    const float* arow  = A + (size_t)(m0 + lo) * K + 2 * hi;
    // B 4x16 f32 layout: VGPR0 -> row k0+2*hi, VGPR1 -> row k0+2*hi+1, N = lane&15
    const float* bbase = B + (size_t)(2 * hi) * N + n0 + lo;

    // prologue: fragments for k0 = 0
    v2f a_cur;
    v2f b_cur[NT];
    a_cur.x = arow[0];
    a_cur.y = arow[1];
    #pragma unroll
    for (int i = 0; i < NT; ++i) {
        b_cur[i].x = bbase[i * 16];
        b_cur[i].y = bbase[i * 16 + N];
    }

    for (int k0 = 4; k0 < K; k0 += 4) {
        // pull upcoming A cachelines toward the WGP (global_prefetch_b8)
        __builtin_prefetch(arow + k0 + 64, 0, 1);
        // issue next k-step's fragment loads before consuming current ones
        v2f a_nxt;
        v2f b_nxt[NT];
        a_nxt.x = arow[k0];
        a_nxt.y = arow[k0 + 1];
        const float* brow = bbase + (size_t)k0 * N;
        #pragma unroll
        for (int i = 0; i < NT; ++i) {
            b_nxt[i].x = brow[i * 16];
            b_nxt[i].y = brow[i * 16 + N];
        }
        #pragma unroll
        for (int i = 0; i < NT; ++i)
            acc[i] = __builtin_amdgcn_wmma_f32_16x16x4_f32(
                false, a_cur, false, b_cur[i], (short)0, acc[i], false, false);
        a_cur = a_nxt;
        #pragma unroll
        for (int i = 0; i < NT; ++i) b_cur[i] = b_nxt[i];
    }
    // epilogue k-step
    #pragma unroll
    for (int i = 0; i < NT; ++i)
        acc[i] = __builtin_amdgcn_wmma_f32_16x16x4_f32(
            false, a_cur, false, b_cur[i], (short)0, acc[i], false, false);

    // C/D 16x16 f32 layout: VGPR r -> M = m0 + r + 8*hi, N = n0 + (lane&15)
    float* drow = D + (size_t)(m0 + 8 * hi) * N + n0 + lo;
    #pragma unroll
    for (int i = 0; i < NT; ++i)
        #pragma unroll
        for (int r = 0; r < 8; ++r)
            drow[(size_t)r * N + i * 16] = acc[i][r];
}

// ---------------------------------------------------------------------------
// Causal depthwise conv (kernel 4) + bias + SiLU, for both directions.
// Backward direction handled by index reversal into shared xz buffer.
// xc layout: [(dir*BATCH + b)*SEQLEN + t][D_INNER], t logical per direction.
// ---------------------------------------------------------------------------
__global__ void conv_silu_kernel(const float* __restrict__ xz,
                                 const float* __restrict__ conv_w,
                                 const float* __restrict__ conv_b,
                                 float* __restrict__ xc)
{
    int idx = blockIdx.x * blockDim.x + threadIdx.x;   // 2^20 elements
    int j = idx & (D_INNER - 1);
    int t = (idx >> 8) & (SEQLEN - 1);
    int b = (idx >> 18) & 1;
    int d = (idx >> 19) & 1;

    float acc = conv_b[j];
    #pragma unroll
    for (int i = 0; i < D_CONV; ++i) {
        int s = t - (D_CONV - 1) + i;                  // causal: w[i]*x[t-3+i]
        if (s >= 0) {
            int sp = d ? (SEQLEN - 1 - s) : s;         // physical index in xz
            acc += conv_w[j * D_CONV + i] *
                   xz[((size_t)(b * SEQLEN + sp)) * XZ_N + j];
        }
    }
    xc[((size_t)((d * BATCH + b) * SEQLEN + t)) * D_INNER + j] = siluf(acc);
}

// ---------------------------------------------------------------------------
// Selective scan. One block per (dir, batch); 1024 threads = 32 waves.
// Thread (q,j): j = tid&255 channel, q = tid>>8 owns k in [64q, 64q+64) states
// kept in registers. y[j] reduced over the 4 q-partials via LDS.
// LDS staging is ping-pong double-buffered -> only 2 barriers per timestep.
// ---------------------------------------------------------------------------
__global__ __launch_bounds__(1024, 1) void scan_kernel(
    const float* __restrict__ proj,
    const float* __restrict__ xc,
    const float* __restrict__ xz,
    const float* __restrict__ A_log,
    const float* __restrict__ Dp,
    float* __restrict__ yg)
{
    __shared__ float sd [2][D_STATE];   // softplus(delta_t)
    __shared__ float sdb[2][D_STATE];   // delta_t * B_t
    __shared__ float sc [2][D_STATE];   // C_t
    __shared__ float su [2][D_INNER];   // u_t = xc[t]
    __shared__ float ypart[1024];

    const int tid   = threadIdx.x;
    const int j     = tid & (D_INNER - 1);
    const int q     = tid >> 8;          // 0..3
    const int kbase = q * 64;
    const int d     = blockIdx.x >> 1;
    const int b     = blockIdx.x & 1;

    const float aj  = -__expf(A_log[j]);   // A[j] = -exp(A_log[j])
    const float dpj = Dp[j];               // valid where j == tid (tid < 256)

    float state[64];
    #pragma unroll
    for (int i = 0; i < 64; ++i) state[i] = 0.0f;

    for (int t = 0; t < SEQLEN; ++t) {
        const int p = t & 1;               // ping-pong staging buffer
        const size_t pr = (size_t)((d * BATCH + b) * SEQLEN + t);

        if (tid < 256) {
            float pd = proj[pr * PROJ_N + tid];
            float pb = proj[pr * PROJ_N + D_STATE + tid];
            float dl = softplusf(pd);
            sd [p][tid] = dl;
            sdb[p][tid] = dl * pb;
        } else if (tid < 512) {
            int k = tid - 256;
            sc[p][k] = proj[pr * PROJ_N + 2 * D_STATE + k];
        } else if (tid < 768) {
            int jj = tid - 512;
            su[p][jj] = xc[pr * D_INNER + jj];
        }
        __syncthreads();                   // staging of step t visible

        const float uu = su[p][j];
        float acc = 0.0f;
        #pragma unroll
        for (int i = 0; i < 64; ++i) {
            float sv = state[i];
            sv = __expf(sd[p][kbase + i] * aj) * sv + sdb[p][kbase + i] * uu;
            state[i] = sv;
            acc = fmaf(sc[p][kbase + i], sv, acc);
        }
        ypart[tid] = acc;
        __syncthreads();                   // partials visible

        if (tid < 256) {
            float y = ypart[tid] + ypart[tid + 256] + ypart[tid + 512] +
                      ypart[tid + 768] + dpj * su[p][tid];
            int tp = d ? (SEQLEN - 1 - t) : t;     // z comes from physical index
            float zv = xz[((size_t)(b * SEQLEN + tp)) * XZ_N + D_INNER + tid];
            yg[pr * D_INNER + tid] = y * siluf(zv);
        }
        // no third barrier: step t+1 stages into buffer p^1, and ypart's next
        // write is ordered behind this step's reads by the two barriers above.
    }
}

// ---------------------------------------------------------------------------
// out_pre[b,t] = 0.5*(yg_fwd[b,t] + yg_bwd[b, L-1-t])  (flip of bwd direction)
// ---------------------------------------------------------------------------
__global__ void combine_kernel(const float* __restrict__ yg,
                               float* __restrict__ yc)
{
    int idx = blockIdx.x * blockDim.x + threadIdx.x;   // 2^19 elements
    int j = idx & (D_INNER - 1);
    int t = (idx >> 8) & (SEQLEN - 1);
    int b = (idx >> 18) & 1;
    size_t rf = (size_t)((0 * BATCH + b) * SEQLEN + t);
    size_t rb = (size_t)((1 * BATCH + b) * SEQLEN + (SEQLEN - 1 - t));
    yc[((size_t)(b * SEQLEN + t)) * D_INNER + j] =
        0.5f * (yg[rf * D_INNER + j] + yg[rb * D_INNER + j]);
}

// ---------------------------------------------------------------------------
extern "C" void kernel_launch(void* const* d_in, const int* in_sizes, int n_in,
                              void* d_out, int out_size, void* d_ws, size_t ws_size,
                              hipStream_t stream)
{
    const float* x      = (const float*)d_in[0];
    const float* W_in   = (const float*)d_in[1];
    const float* conv_w = (const float*)d_in[2];
    const float* conv_b = (const float*)d_in[3];
    const float* W_x    = (const float*)d_in[4];
    const float* A_log  = (const float*)d_in[5];
    const float* Dp     = (const float*)d_in[6];
    const float* W_out  = (const float*)d_in[7];
    float* out = (float*)d_out;

    float* ws   = (float*)d_ws;
    float* xz   = ws;                                   // 2048*512  = 1M floats
    float* xc   = xz   + (size_t)NROWS * XZ_N;          // 4096*256  = 1M floats
    float* proj = xc   + (size_t)2 * NROWS * D_INNER;   // 4096*768  = 3M floats
    float* yg   = proj + (size_t)2 * NROWS * PROJ_N;    // 4096*256  = 1M floats
    float* yc   = yg   + (size_t)2 * NROWS * D_INNER;   // 2048*256  = 512K floats
    (void)ws_size; (void)in_sizes; (void)n_in; (void)out_size;

    // K1: xz = x @ W_in   (2048x128)@(128x512) -- shared by both directions
    gemm_f32_wmma<4, 4><<<dim3(XZ_N / 64, NROWS / 64), 128, 0, stream>>>(
        x, W_in, xz, NROWS, XZ_N, D_MODEL);

    // K2: depthwise causal conv + bias + SiLU (both directions via index flip)
    conv_silu_kernel<<<(2 * NROWS * D_INNER) / 256, 256, 0, stream>>>(
        xz, conv_w, conv_b, xc);

    // K3: proj = [xc_fwd; xc_bwd] @ W_x   (4096x256)@(256x768)
    gemm_f32_wmma<4, 4><<<dim3(PROJ_N / 64, (2 * NROWS) / 64), 128, 0, stream>>>(
        xc, W_x, proj, 2 * NROWS, PROJ_N, D_INNER);

    // K4: selective scan, one block per (dir, batch)
    scan_kernel<<<4, 1024, 0, stream>>>(proj, xc, xz, A_log, Dp, yg);

    // K4b: 0.5*(fwd + flip(bwd))
    combine_kernel<<<(NROWS * D_INNER) / 256, 256, 0, stream>>>(yg, yc);

    // K5: out = yc @ W_out   (2048x256)@(256x128)
    gemm_f32_wmma<4, 4><<<dim3(D_MODEL / 64, NROWS / 64), 128, 0, stream>>>(
        yc, W_out, out, NROWS, D_MODEL, D_INNER);
}